// TransitionModel_36833639531010
// MI455X (gfx1250) — compile-verified
//
#include <hip/hip_runtime.h>
#include <hip/hip_bf16.h>

// ---------------------------------------------------------------------------
// RSSM (PlaNet transition model) for MI455X / gfx1250, wave32, WMMA f32.
//
//  1) rssm_obs_embed: hoisted parallel GEMM  OE[t,b,:] = obs[t,b,:] @ W_bq[:,200:]^T
//     (~37% of FLOPs, no sequential dep). Async global->LDS chunk staging
//     (ASYNCcnt), all 13 weight tiles staged per K-chunk -> 2 barriers/chunk,
//     13 independent WMMA chains for ILP.
//  2) rssm_step (x50, stream-serialized): per-timestep recurrent math.
//     Weight tiles batch-staged (up to 6 at a time, 315 KB LDS of the 320 KB
//     WGP budget) to minimize barrier count on the latency-critical scan.
// ---------------------------------------------------------------------------

typedef __attribute__((ext_vector_type(2))) float v2f;
typedef __attribute__((ext_vector_type(8))) float v8f;

#define T_STEPS 50
#define BATCH   1024
#define SDIM    30
#define ADIM    8
#define BEL     200
#define BELP    208      // padded to 13*16
#define EMB     1024
#define HID     200
#define BLK     256
#define TSLOT   3200     // floats per 200x16 weight tile slot

#define USE_ASYNC_STAGING 1

__device__ __forceinline__ v8f zero8() {
  v8f z = {0.f,0.f,0.f,0.f,0.f,0.f,0.f,0.f};
  return z;
}

__device__ __forceinline__ float sigm(float x)  { return 1.f / (1.f + __expf(-x)); }
__device__ __forceinline__ float softp(float x) { return (x > 20.f) ? x : log1pf(__expf(x)); }

// C(16x16,f32) += A(16xKITER) * Bt(KITERx16) via V_WMMA_F32_16X16X4_F32.
// Fragment layouts per CDNA5 ISA 7.12.2:
//   A: lanes 0-15 -> M=lane, regs {K=k0,k0+1}; lanes 16-31 -> M=lane-16, {K=k0+2,k0+3}
//   B: mirrored (K striped the same way), N = lane&15
//   C: reg r -> M = r + 8*(lane>=16), N = lane&15
template<int KITER, int ASTRIDE>
__device__ __forceinline__ v8f frag_gemm(const float* __restrict__ A,
                                         const float* __restrict__ Bt, v8f c) {
  const int lane = threadIdx.x & 31;
  const int koff = (lane >> 4) << 1;     // 0 or 2
  const int col  = lane & 15;
  const float* ar = A + (size_t)(lane & 15) * ASTRIDE;
#pragma unroll
  for (int k0 = 0; k0 < KITER; k0 += 4) {
    v2f a, b;
    a[0] = ar[k0 + koff];
    a[1] = ar[k0 + koff + 1];
    b[0] = Bt[(k0 + koff) * 16 + col];
    b[1] = Bt[(k0 + koff + 1) * 16 + col];
    c = __builtin_amdgcn_wmma_f32_16x16x4_f32(false, a, false, b, (short)0, c, false, false);
  }
  return c;
}

// ---------------------------------------------------------------------------
// Kernel 1: OE[t][b][n] = obs[t,b,:] @ W_bq[n, 200:1224]^T   (n < 200, pad 208)
// grid = 50*8 blocks; block owns (t, 128 rows). LDS: 64KB A-chunk + 104KB Bt.
// ---------------------------------------------------------------------------
__global__ void __launch_bounds__(BLK, 1)
rssm_obs_embed(const float* __restrict__ obs,
               const float* __restrict__ W_bq,
               float* __restrict__ OE) {
  extern __shared__ float lds[];
  float* sA  = lds;                 // 128*128
  float* sBt = sA + 128 * 128;      // 13 * 128*16

  const int tid   = threadIdx.x;
  const int wave  = tid >> 5;
  const int lane  = tid & 31;
  const int m0    = wave * 16;
  const int col   = lane & 15;
  const int rb    = (lane >> 4) << 3;       // 0 or 8
  const int t     = blockIdx.x >> 3;
  const int gRow0 = (blockIdx.x & 7) * 128;
  const float* ob = obs + ((size_t)t * BATCH + gRow0) * EMB;

  v8f acc[13];
#pragma unroll
  for (int ct = 0; ct < 13; ++ct) acc[ct] = zero8();

  for (int kc = 0; kc < EMB; kc += 128) {
#if USE_ASYNC_STAGING
    // async DMA of the 128x128 f32 chunk straight into LDS (no VGPR bounce),
    // tracked by ASYNCcnt and drained before the barrier.
    for (int j = tid; j < (128 * 128) / 4; j += BLK) {
      const float* g = ob + (size_t)(j >> 5) * EMB + kc + ((j & 31) << 2);
      unsigned l = (unsigned)(size_t)(sA + (j << 2));   // LDS byte offset
      asm volatile("global_load_async_to_lds_b128 %0, %1, off" :: "v"(l), "v"(g) : "memory");
    }
#else
    for (int i = tid; i < 128 * 128; i += BLK) {
      int m = i >> 7, k = i & 127;
      sA[i] = ob[(size_t)m * EMB + kc + k];
    }
#endif
    // stage all 13 weight tiles (Bt[k][n]) for this K-chunk while DMA runs
    for (int i = tid; i < 13 * 2048; i += BLK) {
      int slot = i >> 11;
      int rem  = i & 2047;
      int n = rem >> 7, k = rem & 127;
      int nc = slot * 16 + n;
      float v = (nc < HID) ? W_bq[(size_t)nc * (BEL + EMB) + BEL + kc + k] : 0.f;
      sBt[(slot << 11) + (k << 4) + n] = v;
    }
    // warm caches for the next chunk
    if (kc + 128 < EMB)
      __builtin_prefetch(ob + (size_t)(tid >> 1) * EMB + kc + 128 + ((tid & 1) << 6), 0, 0);
#if USE_ASYNC_STAGING
    asm volatile("s_wait_asynccnt 0x0" ::: "memory");
#endif
    __syncthreads();
    // 13 independent WMMA accumulation chains, back to back (ILP)
#pragma unroll
    for (int ct = 0; ct < 13; ++ct)
      acc[ct] = frag_gemm<128, 128>(sA + (size_t)m0 * 128, sBt + (ct << 11), acc[ct]);
    __syncthreads();
  }
#pragma unroll
  for (int ct = 0; ct < 13; ++ct) {
    int n = ct * 16 + col;
    if (n < HID) {
#pragma unroll
      for (int r = 0; r < 8; ++r)
        OE[((size_t)t * BATCH + gRow0 + m0 + rb + r) * BELP + n] = acc[ct][r];
    }
  }
}

// ---------------------------------------------------------------------------
// Kernel 2: copy initial carries into workspace ping-pong buffers.
// ---------------------------------------------------------------------------
__global__ void rssm_init(const float* __restrict__ pb, const float* __restrict__ ps,
                          float* __restrict__ belC, float* __restrict__ stC) {
  int i = blockIdx.x * BLK + threadIdx.x;
  if (i < BATCH * BEL)  belC[i] = pb[i];
  if (i < BATCH * SDIM) stC[i]  = ps[i];
}

// ---------------------------------------------------------------------------
// Kernel 3: one recurrent timestep. 8 blocks x 256 threads (8 waves).
// ---------------------------------------------------------------------------
struct StepP {
  const float *actions, *nonterm, *noiseP, *noiseQ;
  const float *W_sa, *b_sa, *W_ih, *b_ih, *W_hh, *b_hh;
  const float *W_bp, *b_bp, *W_sp, *b_sp, *W_bq, *b_bq, *W_sq, *b_sq;
  const float *belPrev, *stPrev;
  float *belNext, *stNext;
  const float *OE;
  float *oBel, *oPSt, *oPM, *oPS, *oQSt, *oQM, *oQS;
  int t;
};

__global__ void __launch_bounds__(BLK, 1) rssm_step(StepP p) {
  extern __shared__ float lds[];
  float* sH     = lds;                   // 128*208  (hidden -> hp -> hq)
  float* sBel   = sH + 128 * BELP;       // 128*208  (new belief)
  float* sChunk = sBel + 128 * BELP;     // 128*64   (X input / head outputs)
  float* sBt    = sChunk + 128 * 64;     // 6 * 200*16 weight tile slots

  const int tid   = threadIdx.x;
  const int wave  = tid >> 5;
  const int lane  = tid & 31;
  const int m0    = wave * 16;
  const int col   = lane & 15;
  const int rb    = (lane >> 4) << 3;
  const int gRow0 = blockIdx.x * 128;
  const int t     = p.t;
  const size_t tb = (size_t)t * BATCH + gRow0;

  // --- phase 1: X = [post_state * nt, action] (pad K 38->64) + stage all
  // --- 13 W_sa tiles (13*40*16 floats fit in the slot area) --------------
  {
    const float* act = p.actions + tb * ADIM;
    const float* st  = p.stPrev + (size_t)gRow0 * SDIM;
    const float* nt  = p.nonterm + tb;
    for (int i = tid; i < 128 * 64; i += BLK) {
      int m = i >> 6, k = i & 63;
      float v = 0.f;
      if (k < SDIM)             v = st[m * SDIM + k] * nt[m];
      else if (k < SDIM + ADIM) v = act[m * ADIM + (k - SDIM)];
      sChunk[i] = v;
    }
    for (int i = tid; i < 13 * 640; i += BLK) {
      int slot = i / 640;
      int rem  = i - slot * 640;
      int n = rem / 40, k = rem - n * 40;
      int nc = slot * 16 + n;
      float v = (nc < BEL && k < SDIM + ADIM) ? p.W_sa[(size_t)nc * (SDIM + ADIM) + k] : 0.f;
      sBt[slot * 640 + k * 16 + n] = v;
    }
  }
  // warm GRU weights (L2/WGP$) while phase 2 computes
  __builtin_prefetch(p.W_ih + (size_t)tid * 400, 0, 0);
  __builtin_prefetch(p.W_hh + (size_t)tid * 400, 0, 0);
  __syncthreads();

  // --- phase 2: hidden = relu(X @ W_sa^T + b_sa) -> sH (single barrier) --
  for (int ct = 0; ct < 13; ++ct) {
    v8f c = frag_gemm<40, 64>(sChunk + (size_t)m0 * 64, sBt + ct * 640, zero8());
    int n = ct * 16 + col;
    if (n < BEL) {
      float bb = p.b_sa[n];
#pragma unroll
      for (int r = 0; r < 8; ++r) {
        float v = c[r] + bb;
        sH[(m0 + rb + r) * BELP + n] = v > 0.f ? v : 0.f;
      }
    }
  }
  __syncthreads();

  // --- phase 3: GRU (6 gate tiles staged at once; 2 barriers per ct) -----
  const float* bp0 = p.belPrev + (size_t)gRow0 * BEL;
  for (int ct = 0; ct < 13; ++ct) {
    for (int i = tid; i < 6 * TSLOT; i += BLK) {
      int slot = i / TSLOT;
      int rem  = i - slot * TSLOT;
      int n = rem / BEL, k = rem - n * BEL;
      const float* W = (slot < 3) ? p.W_ih : p.W_hh;
      int nc = ct * 16 + n;
      float v = (nc < BEL) ? W[(size_t)((slot % 3) * BEL + nc) * BEL + k] : 0.f;
      sBt[slot * TSLOT + k * 16 + n] = v;
    }
    __syncthreads();
    v8f g[6];
#pragma unroll
    for (int gi = 0; gi < 6; ++gi) {
      if (gi < 3) g[gi] = frag_gemm<200, BELP>(sH + (size_t)m0 * BELP, sBt + gi * TSLOT, zero8());
      else        g[gi] = frag_gemm<200, BEL>(bp0 + (size_t)m0 * BEL, sBt + gi * TSLOT, zero8());
    }
    __syncthreads();
    int cc = ct * 16 + col;
    if (cc < BEL) {
      float bir = p.b_ih[cc],           bhr = p.b_hh[cc];
      float biz = p.b_ih[BEL + cc],     bhz = p.b_hh[BEL + cc];
      float bin = p.b_ih[2 * BEL + cc], bhn = p.b_hh[2 * BEL + cc];
#pragma unroll
      for (int r = 0; r < 8; ++r) {
        int m = m0 + rb + r;
        float rg = sigm(g[0][r] + bir + g[3][r] + bhr);
        float zg = sigm(g[1][r] + biz + g[4][r] + bhz);
        float ng = tanhf(g[2][r] + bin + rg * (g[5][r] + bhn));
        float bprev = bp0[(size_t)m * BEL + cc];
        float bn = (1.f - zg) * ng + zg * bprev;
        sBel[m * BELP + cc] = bn;
        p.oBel[(tb + m) * BEL + cc]               = bn;
        p.belNext[(size_t)(gRow0 + m) * BEL + cc] = bn;
      }
    }
  }
  __syncthreads();

  // --- phase 4: hp = relu(belief @ W_bp^T + b_bp) -> sH (groups of 6) ----
  for (int g0 = 0; g0 < 13; g0 += 6) {
    int cnt = (13 - g0) < 6 ? (13 - g0) : 6;
    for (int i = tid; i < cnt * TSLOT; i += BLK) {
      int slot = i / TSLOT;
      int rem  = i - slot * TSLOT;
      int n = rem / BEL, k = rem - n * BEL;
      int nc = (g0 + slot) * 16 + n;
      float v = (nc < HID) ? p.W_bp[(size_t)nc * BEL + k] : 0.f;
      sBt[slot * TSLOT + k * 16 + n] = v;
    }
    __syncthreads();
    for (int j = 0; j < cnt; ++j) {
      v8f c = frag_gemm<200, BELP>(sBel + (size_t)m0 * BELP, sBt + j * TSLOT, zero8());
      int n = (g0 + j) * 16 + col;
      if (n < HID) {
        float bb = p.b_bp[n];
#pragma unroll
        for (int r = 0; r < 8; ++r) {
          float v = c[r] + bb;
          sH[(m0 + rb + r) * BELP + n] = v > 0.f ? v : 0.f;
        }
      }
    }
    __syncthreads();
  }

  // --- phase 5: prior head (N=60, 4 tiles in one stage) ------------------
  {
    for (int i = tid; i < 4 * TSLOT; i += BLK) {
      int slot = i / TSLOT;
      int rem  = i - slot * TSLOT;
      int n = rem / BEL, k = rem - n * BEL;
      int nc = slot * 16 + n;
      float v = (nc < 2 * SDIM) ? p.W_sp[(size_t)nc * HID + k] : 0.f;
      sBt[slot * TSLOT + k * 16 + n] = v;
    }
    __syncthreads();
    for (int j = 0; j < 4; ++j) {
      v8f c = frag_gemm<200, BELP>(sH + (size_t)m0 * BELP, sBt + j * TSLOT, zero8());
      int n = j * 16 + col;
      if (n < 2 * SDIM) {
        float bb = p.b_sp[n];
#pragma unroll
        for (int r = 0; r < 8; ++r) sChunk[(m0 + rb + r) * 64 + n] = c[r] + bb;
      }
    }
    __syncthreads();
    const float* nz = p.noiseP + tb * SDIM;
    for (int i = tid; i < 128 * SDIM; i += BLK) {
      int m = i / SDIM, cc = i - m * SDIM;
      size_t o = (tb + m) * SDIM + cc;
      float mo = sChunk[m * 64 + cc];
      float sd = softp(sChunk[m * 64 + SDIM + cc]) + 0.1f;
      p.oPM[o]  = mo;
      p.oPS[o]  = sd;
      p.oPSt[o] = mo + sd * nz[i];
    }
    __syncthreads();
  }

  // --- phase 6: hq = relu(belief @ W_bq[:, :200]^T + OE + b_bq) -> sH ----
  for (int g0 = 0; g0 < 13; g0 += 6) {
    int cnt = (13 - g0) < 6 ? (13 - g0) : 6;
    for (int i = tid; i < cnt * TSLOT; i += BLK) {
      int slot = i / TSLOT;
      int rem  = i - slot * TSLOT;
      int n = rem / BEL, k = rem - n * BEL;
      int nc = (g0 + slot) * 16 + n;
      float v = (nc < HID) ? p.W_bq[(size_t)nc * (BEL + EMB) + k] : 0.f;
      sBt[slot * TSLOT + k * 16 + n] = v;
    }
    __syncthreads();
    for (int j = 0; j < cnt; ++j) {
      v8f c = frag_gemm<200, BELP>(sBel + (size_t)m0 * BELP, sBt + j * TSLOT, zero8());
      int n = (g0 + j) * 16 + col;
      if (n < HID) {
        float bb = p.b_bq[n];
#pragma unroll
        for (int r = 0; r < 8; ++r) {
          int m = m0 + rb + r;
          float v = c[r] + bb + p.OE[(tb + m) * BELP + n];
          sH[m * BELP + n] = v > 0.f ? v : 0.f;
        }
      }
    }
    __syncthreads();
  }

  // --- phase 7: posterior head (N=60) + outputs + carry ------------------
  {
    for (int i = tid; i < 4 * TSLOT; i += BLK) {
      int slot = i / TSLOT;
      int rem  = i - slot * TSLOT;
      int n = rem / BEL, k = rem - n * BEL;
      int nc = slot * 16 + n;
      float v = (nc < 2 * SDIM) ? p.W_sq[(size_t)nc * HID + k] : 0.f;
      sBt[slot * TSLOT + k * 16 + n] = v;
    }
    __syncthreads();
    for (int j = 0; j < 4; ++j) {
      v8f c = frag_gemm<200, BELP>(sH + (size_t)m0 * BELP, sBt + j * TSLOT, zero8());
      int n = j * 16 + col;
      if (n < 2 * SDIM) {
        float bb = p.b_sq[n];
#pragma unroll
        for (int r = 0; r < 8; ++r) sChunk[(m0 + rb + r) * 64 + n] = c[r] + bb;
      }
    }
    __syncthreads();
    const float* nz = p.noiseQ + tb * SDIM;
    for (int i = tid; i < 128 * SDIM; i += BLK) {
      int m = i / SDIM, cc = i - m * SDIM;
      size_t o = (tb + m) * SDIM + cc;
      float mo = sChunk[m * 64 + cc];
      float sd = softp(sChunk[m * 64 + SDIM + cc]) + 0.1f;
      float st = mo + sd * nz[i];
      p.oQM[o]  = mo;
      p.oQS[o]  = sd;
      p.oQSt[o] = st;
      p.stNext[(size_t)(gRow0 + m) * SDIM + cc] = st;
    }
  }
}

// ---------------------------------------------------------------------------
extern "C" void kernel_launch(void* const* d_in, const int* in_sizes, int n_in,
                              void* d_out, int out_size, void* d_ws, size_t ws_size,
                              hipStream_t stream) {
  const float* prev_state   = (const float*)d_in[0];
  const float* actions      = (const float*)d_in[1];
  const float* prev_belief  = (const float*)d_in[2];
  const float* observations = (const float*)d_in[3];
  const float* nonterminals = (const float*)d_in[4];
  const float* noise_prior  = (const float*)d_in[5];
  const float* noise_post   = (const float*)d_in[6];
  const float* W_sa = (const float*)d_in[7];
  const float* b_sa = (const float*)d_in[8];
  const float* W_ih = (const float*)d_in[9];
  const float* b_ih = (const float*)d_in[10];
  const float* W_hh = (const float*)d_in[11];
  const float* b_hh = (const float*)d_in[12];
  const float* W_bp = (const float*)d_in[13];
  const float* b_bp = (const float*)d_in[14];
  const float* W_sp = (const float*)d_in[15];
  const float* b_sp = (const float*)d_in[16];
  const float* W_bq = (const float*)d_in[17];
  const float* b_bq = (const float*)d_in[18];
  const float* W_sq = (const float*)d_in[19];
  const float* b_sq = (const float*)d_in[20];

  float* out = (float*)d_out;
  float* ws  = (float*)d_ws;

  // workspace: OE[50][1024][208] + belief/state ping-pong carries (~44.5 MB)
  float* OE   = ws;
  float* belC = OE + (size_t)T_STEPS * BATCH * BELP;
  float* stC  = belC + (size_t)2 * BATCH * BEL;

  const size_t TB = (size_t)T_STEPS * BATCH;
  float* oBel = out;
  float* oPSt = oBel + TB * BEL;
  float* oPM  = oPSt + TB * SDIM;
  float* oPS  = oPM  + TB * SDIM;
  float* oQSt = oPS  + TB * SDIM;
  float* oQM  = oQSt + TB * SDIM;
  float* oQS  = oQM  + TB * SDIM;

  size_t ldsPre  = (size_t)(128 * 128 + 13 * 128 * 16) * sizeof(float);
  size_t ldsStep = (size_t)(128 * BELP * 2 + 128 * 64 + 6 * TSLOT) * sizeof(float);

  rssm_obs_embed<<<T_STEPS * 8, BLK, ldsPre, stream>>>(observations, W_bq, OE);
  rssm_init<<<(BATCH * BEL + BLK - 1) / BLK, BLK, 0, stream>>>(prev_belief, prev_state,
                                                              belC, stC);

  for (int t = 0; t < T_STEPS; ++t) {
    StepP p;
    p.actions = actions;  p.nonterm = nonterminals;
    p.noiseP  = noise_prior;  p.noiseQ = noise_post;
    p.W_sa = W_sa; p.b_sa = b_sa;
    p.W_ih = W_ih; p.b_ih = b_ih;
    p.W_hh = W_hh; p.b_hh = b_hh;
    p.W_bp = W_bp; p.b_bp = b_bp;
    p.W_sp = W_sp; p.b_sp = b_sp;
    p.W_bq = W_bq; p.b_bq = b_bq;
    p.W_sq = W_sq; p.b_sq = b_sq;
    p.belPrev = belC + (size_t)(t & 1) * BATCH * BEL;
    p.belNext = belC + (size_t)((t + 1) & 1) * BATCH * BEL;
    p.stPrev  = stC + (size_t)(t & 1) * BATCH * SDIM;
    p.stNext  = stC + (size_t)((t + 1) & 1) * BATCH * SDIM;
    p.OE = OE;
    p.oBel = oBel; p.oPSt = oPSt; p.oPM = oPM; p.oPS = oPS;
    p.oQSt = oQSt; p.oQM = oQM; p.oQS = oQS;
    p.t = t;
    rssm_step<<<8, BLK, ldsStep, stream>>>(p);
  }
}